// CustomCrossAttentionExt_38646115729794
// MI455X (gfx1250) — compile-verified
//
#include <hip/hip_runtime.h>
#include <math.h>

#define BB 8
#define NN 4096
#define JJ 308
#define JPAD 320
#define QDIM 640
#define CDIM 768
#define NH 8
#define DHEAD 80
#define ATT_SCALE 0.11180339887498948f   // 80^-0.5

typedef float v2f __attribute__((ext_vector_type(2)));
typedef float v8f __attribute__((ext_vector_type(8)));

// ---- workspace layout (float offsets) ----
#define OFF_PE  0L                                   // 8*640
#define OFF_SC  5120L                                // S1,S2,cntJ,sigma
#define OFF_EIN 5376L                                // B*J*768 (ein -> kin in place)
#define OFF_H1  (OFF_EIN + 1892352L)                 // B*J*384
#define OFF_KT  (OFF_H1 + 946176L)                   // B*H*80*JPAD (k transposed)
#define OFF_V   (OFF_KT + 1638400L)                  // B*J*640
#define OFF_Q   (OFF_V + 1576960L)                   // B*N*640 (q -> attn-out in place)

__device__ __forceinline__ v8f wmma4(v2f a, v2f b, v8f c) {
  // D = A(16x4 f32) x B(4x16 f32) + C, fp32 accumulate on the matrix pipe
  return __builtin_amdgcn_wmma_f32_16x16x4_f32(false, a, false, b, (short)0, c, false, false);
}
__device__ __forceinline__ v8f zero8() {
  v8f z = {0.f, 0.f, 0.f, 0.f, 0.f, 0.f, 0.f, 0.f};
  return z;
}
__device__ __forceinline__ float gelu_f(float x) {
  return 0.5f * x * (1.0f + erff(x * 0.70710678118654752f));
}

// 16-row GEMM, A resident in LDS (row stride lda, lda%64==4 to spread banks)
__device__ __forceinline__ v8f mm_ldsA(const float* As, int lda,
                                       const float* __restrict__ Bw, int ldb,
                                       int n0, int K, int lane) {
  const int half = lane >> 4, ln = lane & 15;
  v8f acc = zero8();
  const float* ar = As + ln * lda + half * 2;
  const float* bc = Bw + (long)(half * 2) * ldb + n0 + ln;
  for (int k = 0; k < K; k += 4) {
    v2f a, b;
    a.x = ar[k];
    a.y = ar[k + 1];
    b.x = bc[(long)k * ldb];
    b.y = bc[(long)(k + 1) * ldb];
    acc = wmma4(a, b, acc);
  }
  return acc;
}

// 16-row GEMM streaming A from global through a double-buffered 16x64 LDS chunk.
// Register-staged (b128) prefetch of chunk ci+1 is issued before computing chunk ci,
// hiding global latency behind the WMMA burst. Tile t sits at col n0base + t*128.
template <int NT>
__device__ __forceinline__ void mm_streamA(const float* __restrict__ A, long arow0, int lda,
                                           int nvalid, const float* __restrict__ Bw, int ldb,
                                           int K, int n0base, v8f* acc,
                                           float* chunk /* LDS 2*16*68 */) {
  const int tid = threadIdx.x;
  const int lane = tid & 31, half = lane >> 4, ln = lane & 15;
  const int r0 = tid >> 4, c0 = (tid & 15) * 4;
  const int nch = K / 64;
  float4 st;
  if (r0 < nvalid) st = *(const float4*)(A + arow0 + (long)r0 * lda + c0);
  else st = make_float4(0.f, 0.f, 0.f, 0.f);
  for (int ci = 0; ci < nch; ++ci) {
    float* cur = chunk + (ci & 1) * (16 * 68);
    __syncthreads();                       // buffer ci&1 last read 2 iterations ago
    *(float4*)(cur + r0 * 68 + c0) = st;
    if (ci + 1 < nch) {
      if (r0 < nvalid) st = *(const float4*)(A + arow0 + (long)r0 * lda + (ci + 1) * 64 + c0);
      else st = make_float4(0.f, 0.f, 0.f, 0.f);
    }
    __syncthreads();
    const float* ar = cur + ln * 68 + half * 2;
    const int kc = ci * 64;
#pragma unroll
    for (int t = 0; t < NT; ++t) {
      const float* bc = Bw + (long)(kc + half * 2) * ldb + n0base + t * 128 + ln;
      v8f a8 = acc[t];
#pragma unroll
      for (int k = 0; k < 64; k += 4) {
        v2f a, b;
        a.x = ar[k];
        a.y = ar[k + 1];
        b.x = bc[(long)k * ldb];
        b.y = bc[(long)(k + 1) * ldb];
        a8 = wmma4(a, b, a8);
      }
      acc[t] = a8;
    }
  }
  __syncthreads();
}

// ---------------- kernels ----------------

__global__ void k_init(float* ws) {
  if (threadIdx.x < 8) ws[OFF_SC + threadIdx.x] = 0.0f;
}

__global__ __launch_bounds__(320) void k_pe(const float* __restrict__ prog,
                                            const float* __restrict__ w1,
                                            const float* __restrict__ b1,
                                            const float* __restrict__ w2,
                                            const float* __restrict__ b2, float* ws) {
  __shared__ float h[320];
  int b = blockIdx.x, t = threadIdx.x;
  float p = prog[b];
  h[t] = fmaxf(0.f, p * w1[t] + b1[t]);
  __syncthreads();
  for (int o = t; o < 640; o += 320) {
    float s = b2[o];
    for (int j = 0; j < 320; ++j) s += h[j] * w2[j * 640 + o];
    ws[OFF_PE + b * 640 + o] = s;
  }
}

__global__ void k_ein(const float* __restrict__ embs, const int* __restrict__ ct,
                      const float* __restrict__ tt_emb, float* ws) {
  long idx4 = (long)blockIdx.x * 256 + threadIdx.x;
  long total4 = (long)BB * JJ * CDIM / 4;
  if (idx4 >= total4) return;
  long idx = idx4 * 4;
  int c = (int)(idx % CDIM);
  long bj = idx / CDIM;
  int t = ct[bj];
  if (t < 0) t = 0;
  float4 e = *(const float4*)(embs + idx);
  float4 tt = *(const float4*)(tt_emb + (long)t * CDIM + c);
  float4 o = make_float4(e.x + tt.x, e.y + tt.y, e.z + tt.z, e.w + tt.w);
  *(float4*)(ws + OFF_EIN + idx) = o;
}

// h1 = gelu(ein @ tt_w1 + tt_b1)
__global__ __launch_bounds__(256) void k_g1(const float* __restrict__ w1,
                                            const float* __restrict__ b1, float* ws) {
  __shared__ float chunk[2 * 16 * 68];
  int bg = blockIdx.x / 20, tj = blockIdx.x % 20;
  int j0 = tj * 16;
  int nvalid = JJ - j0; if (nvalid > 16) nvalid = 16;
  int lane = threadIdx.x & 31, wave = threadIdx.x >> 5;
  int half = lane >> 4, ln = lane & 15;
  v8f acc[3];
#pragma unroll
  for (int t = 0; t < 3; ++t) acc[t] = zero8();
  mm_streamA<3>(ws + OFF_EIN, ((long)bg * JJ + j0) * CDIM, CDIM, nvalid, w1, 384, CDIM,
                wave * 16, acc, chunk);
#pragma unroll
  for (int t = 0; t < 3; ++t) {
    int n = wave * 16 + t * 128 + ln;
#pragma unroll
    for (int r = 0; r < 8; ++r) {
      int m = r + half * 8, j = j0 + m;
      if (j < JJ) ws[OFF_H1 + ((long)bg * JJ + j) * 384 + n] = gelu_f(acc[t][r] + b1[n]);
    }
  }
}

// kin = embs*(1-gA) + ein*gA + (h1@tt_w2 + tt_b2)*gB, written over ein
__global__ __launch_bounds__(256) void k_g2(const float* __restrict__ embs,
                                            const float* __restrict__ w2,
                                            const float* __restrict__ b2,
                                            const float* __restrict__ gA,
                                            const float* __restrict__ gB, float* ws) {
  __shared__ float chunk[2 * 16 * 68];
  int bg = blockIdx.x / 20, tj = blockIdx.x % 20;
  int j0 = tj * 16;
  int nvalid = JJ - j0; if (nvalid > 16) nvalid = 16;
  int lane = threadIdx.x & 31, wave = threadIdx.x >> 5;
  int half = lane >> 4, ln = lane & 15;
  v8f acc[6];
#pragma unroll
  for (int t = 0; t < 6; ++t) acc[t] = zero8();
  mm_streamA<6>(ws + OFF_H1, ((long)bg * JJ + j0) * 384, 384, nvalid, w2, CDIM, 384,
                wave * 16, acc, chunk);
#pragma unroll
  for (int t = 0; t < 6; ++t) {
    int n = wave * 16 + t * 128 + ln;
    float ga = gA[n], gb = gB[n], bb = b2[n];
#pragma unroll
    for (int r = 0; r < 8; ++r) {
      int m = r + half * 8, j = j0 + m;
      if (j < JJ) {
        long ridx = ((long)bg * JJ + j) * CDIM + n;
        float e = embs[ridx];
        float ei = ws[OFF_EIN + ridx];
        float tm = acc[t][r] + bb;
        ws[OFF_EIN + ridx] = e * (1.f - ga) + ei * ga + tm * gb;
      }
    }
  }
}

// k = kin @ Wk, stored transposed kT[(b,h,d), j] with J padded to JPAD
__global__ __launch_bounds__(256) void k_g3(const float* __restrict__ Wk, float* ws) {
  __shared__ float chunk[2 * 16 * 68];
  int bg = blockIdx.x / 20, tj = blockIdx.x % 20;
  int j0 = tj * 16;
  int nvalid = JJ - j0; if (nvalid > 16) nvalid = 16;
  int lane = threadIdx.x & 31, wave = threadIdx.x >> 5;
  int half = lane >> 4, ln = lane & 15;
  v8f acc[5];
#pragma unroll
  for (int t = 0; t < 5; ++t) acc[t] = zero8();
  mm_streamA<5>(ws + OFF_EIN, ((long)bg * JJ + j0) * CDIM, CDIM, nvalid, Wk, 640, CDIM,
                wave * 16, acc, chunk);
#pragma unroll
  for (int t = 0; t < 5; ++t) {
    int n = wave * 16 + t * 128 + ln;
    int hh = n / 80, dd = n - hh * 80;
#pragma unroll
    for (int r = 0; r < 8; ++r) {
      int m = r + half * 8, j = j0 + m;
      if (j < JJ)
        ws[OFF_KT + ((long)(bg * NH + hh) * DHEAD + dd) * JPAD + j] = acc[t][r];
    }
  }
}

// v = embs @ Wv
__global__ __launch_bounds__(256) void k_g4(const float* __restrict__ embs,
                                            const float* __restrict__ Wv, float* ws) {
  __shared__ float chunk[2 * 16 * 68];
  int bg = blockIdx.x / 20, tj = blockIdx.x % 20;
  int j0 = tj * 16;
  int nvalid = JJ - j0; if (nvalid > 16) nvalid = 16;
  int lane = threadIdx.x & 31, wave = threadIdx.x >> 5;
  int half = lane >> 4, ln = lane & 15;
  v8f acc[5];
#pragma unroll
  for (int t = 0; t < 5; ++t) acc[t] = zero8();
  mm_streamA<5>(embs, ((long)bg * JJ + j0) * CDIM, CDIM, nvalid, Wv, 640, CDIM,
                wave * 16, acc, chunk);
#pragma unroll
  for (int t = 0; t < 5; ++t) {
    int n = wave * 16 + t * 128 + ln;
#pragma unroll
    for (int r = 0; r < 8; ++r) {
      int m = r + half * 8, j = j0 + m;
      if (j < JJ) ws[OFF_V + ((long)bg * JJ + j) * 640 + n] = acc[t][r];
    }
  }
}

// fused: a0=x+pe -> h2=gelu(a0@pm_w1+b1) -> pm -> qin -> q=qin@Wq
__global__ __launch_bounds__(256) void k_qp(const float* __restrict__ x,
                                            const float* __restrict__ pm_w1,
                                            const float* __restrict__ pm_b1,
                                            const float* __restrict__ pm_w2,
                                            const float* __restrict__ pm_b2,
                                            const float* __restrict__ pg_gA,
                                            const float* __restrict__ pg_gB,
                                            const float* __restrict__ Wq, float* ws) {
  __shared__ float a0[16 * 644];
  __shared__ float h2[16 * 324];
  int b = blockIdx.x >> 8;
  int i0 = (blockIdx.x & 255) * 16;
  int tid = threadIdx.x, wave = tid >> 5, lane = tid & 31, half = lane >> 4, ln = lane & 15;
  const float* pe = ws + OFF_PE + b * 640;
  for (int idx = tid; idx < 16 * 160; idx += 256) {
    int r = idx / 160, c4 = (idx - r * 160) * 4;
    float4 xv = *(const float4*)(x + ((long)b * NN + i0 + r) * 640 + c4);
    float4 pv = *(const float4*)(pe + c4);
    *(float4*)(a0 + r * 644 + c4) =
        make_float4(xv.x + pv.x, xv.y + pv.y, xv.z + pv.z, xv.w + pv.w);
  }
  __syncthreads();
  for (int t = wave; t < 20; t += 8) {          // GEMM1 -> h2
    int n0 = t * 16;
    v8f acc = mm_ldsA(a0, 644, pm_w1, 320, n0, 640, lane);
#pragma unroll
    for (int r = 0; r < 8; ++r)
      h2[(r + half * 8) * 324 + n0 + ln] = gelu_f(acc[r] + pm_b1[n0 + ln]);
  }
  __syncthreads();
  for (int t = wave; t < 40; t += 8) {          // GEMM2 -> qin in a0
    int n0 = t * 16;
    v8f acc = mm_ldsA(h2, 324, pm_w2, 640, n0, 320, lane);
    int n = n0 + ln;
    float pev = pe[n], ga = pg_gA[n], gb = pg_gB[n], b2 = pm_b2[n];
#pragma unroll
    for (int r = 0; r < 8; ++r) {
      int m = r + half * 8;
      float pmv = acc[r] + b2;
      float a0v = a0[m * 644 + n];
      a0[m * 644 + n] = (a0v - pev) + pev * ga + pmv * gb;
    }
  }
  __syncthreads();
  for (int t = wave; t < 40; t += 8) {          // GEMM3 -> q
    int n0 = t * 16;
    v8f acc = mm_ldsA(a0, 644, Wq, 640, n0, 640, lane);
#pragma unroll
    for (int r = 0; r < 8; ++r) {
      int m = r + half * 8;
      ws[OFF_Q + ((long)b * NN + i0 + m) * 640 + n0 + ln] = acc[r];
    }
  }
}

// per (b,h): Q-gram via LDS-chunked WMMA (+fused qsum), K-gram (VALU from LDS mask),
// then S1 += qsum.ksum and S2 += <Qgram,Kgram>
__global__ __launch_bounds__(256) void k_gram(const int* __restrict__ ct, float* ws) {
  __shared__ float qc[64 * 84];
  __shared__ float Kg[80 * 84];
  __shared__ float maskv[JPAD];
  __shared__ float qs[80], ks[80];
  __shared__ float red[256];
  int b = blockIdx.x >> 3, h = blockIdx.x & 7;
  int tid = threadIdx.x;
  int wave = tid >> 5, lane = tid & 31, half = lane >> 4, ln = lane & 15;
  const float* kT = ws + OFF_KT + (long)(b * NH + h) * DHEAD * JPAD;
  const float* qb = ws + OFF_Q + (long)b * NN * 640 + h * 80;
  for (int j = tid; j < JPAD; j += 256)
    maskv[j] = (j < JJ && ct[b * JJ + j] >= 0) ? 1.0f : 0.0f;

  // --- Q-gram (WMMA out of LDS chunks) + qsum fused ---
  v8f ga[4];
#pragma unroll
  for (int u = 0; u < 4; ++u) ga[u] = zero8();
  float qsum_l = 0.f;
  for (int cb = 0; cb < NN; cb += 64) {
    __syncthreads();
    for (int idx = tid; idx < 64 * 20; idx += 256) {
      int r = idx / 20, c4 = (idx - r * 20) * 4;
      *(float4*)(qc + r * 84 + c4) = *(const float4*)(qb + (long)(cb + r) * 640 + c4);
    }
    __syncthreads();
#pragma unroll
    for (int u = 0; u < 4; ++u) {
      int t = wave + u * 8;
      if (t < 25) {
        int tm = t / 5, tn = t - tm * 5;
        v8f a8 = ga[u];
#pragma unroll
        for (int k = 0; k < 64; k += 4) {
          int ka = k + half * 2;
          v2f a, bb;
          a.x = qc[ka * 84 + tm * 16 + ln];
          a.y = qc[(ka + 1) * 84 + tm * 16 + ln];
          bb.x = qc[ka * 84 + tn * 16 + ln];
          bb.y = qc[(ka + 1) * 84 + tn * 16 + ln];
          a8 = wmma4(a, bb, a8);
        }
        ga[u] = a8;
      }
    }
    if (tid < 80) {
      float s = 0.f;
      for (int r = 0; r < 64; ++r) s += qc[r * 84 + tid];
      qsum_l += s;
    }
  }
  __syncthreads();
  if (tid < 80) qs[tid] = qsum_l;

  // --- K-gram + ksum + cnt ---
  for (int e = tid; e < 6400; e += 256) {
    int d1 = e / 80, d2 = e - d1 * 80;
    const float* r1 = kT + d1 * JPAD;
    const float* r2 = kT + d2 * JPAD;
    float s = 0.f;
    for (int j = 0; j < JJ; ++j) s += maskv[j] * r1[j] * r2[j];
    Kg[d1 * 84 + d2] = s;
  }
  if (tid < 80) {
    const float* r1 = kT + tid * JPAD;
    float s = 0.f;
    for (int j = 0; j < JJ; ++j) s += maskv[j] * r1[j];
    ks[tid] = s;
  } else if (tid == 80 && h == 0) {
    float c = 0.f;
    for (int j = 0; j < JJ; ++j) c += maskv[j];
    atomicAdd(ws + OFF_SC + 2, c);
  }
  __syncthreads();
  if (tid == 0) {
    float s = 0.f;
    for (int d = 0; d < 80; ++d) s += qs[d] * ks[d];
    atomicAdd(ws + OFF_SC + 0, s);
  }
  // --- Frobenius <Qgram, Kgram> ---
  float s2 = 0.f;
#pragma unroll
  for (int u = 0; u < 4; ++u) {
    int t = wave + u * 8;
    if (t < 25) {
      int tm = t / 5, tn = t - tm * 5;
#pragma unroll
      for (int r = 0; r < 8; ++r)
        s2 += ga[u][r] * Kg[(tm * 16 + r + half * 8) * 84 + tn * 16 + ln];
    }
  }
  red[tid] = s2;
  __syncthreads();
  if (tid == 0) {
    float s = 0.f;
    for (int i = 0; i < 256; ++i) s += red[i];
    atomicAdd(ws + OFF_SC + 1, s);
  }
}

__global__ void k_scal(const float* __restrict__ strength, float* ws) {
  float S1 = ws[OFF_SC + 0], S2 = ws[OFF_SC + 1], cj = ws[OFF_SC + 2];
  float cnt = cj * (float)(NH * NN);
  float mean = S1 / cnt;
  float var = (S2 - cnt * mean * mean) / (cnt - 1.0f);
  ws[OFF_SC + 3] = sqrtf(fmaxf(var, 0.f)) * strength[0];
}

// attention: sim = q@kT (+mask epilogue), softmax, attn@v -> overwrite q region
__global__ __launch_bounds__(256) void k_attn(const float* __restrict__ cam,
                                              const int* __restrict__ ct, float* ws) {
  __shared__ float S[16 * 324];
  __shared__ float qS[16 * 84];
  __shared__ float red[16 * 17];
  __shared__ float rowmax[16], rowsum[16];
  int b = blockIdx.x >> 11;
  int rem = blockIdx.x & 2047;
  int h = rem >> 8;
  int i0 = (rem & 255) * 16;
  int tid = threadIdx.x, wave = tid >> 5, lane = tid & 31, half = lane >> 4, ln = lane & 15;
  float sigma = ws[OFF_SC + 3];
  const float* qb = ws + OFF_Q + ((long)b * NN + i0) * 640 + h * 80;
  const float* kT = ws + OFF_KT + (long)(b * NH + h) * DHEAD * JPAD;
  // stage 16x80 q tile (coalesced b128 -> LDS)
  for (int idx = tid; idx < 16 * 20; idx += 256) {
    int r = idx / 20, c4 = (idx - r * 20) * 4;
    *(float4*)(qS + r * 84 + c4) = *(const float4*)(qb + (long)r * 640 + c4);
  }
  __syncthreads();
  for (int t = wave; t < 20; t += 8) {
    int j0 = t * 16;
    v8f acc = zero8();
#pragma unroll
    for (int k = 0; k < DHEAD; k += 4) {
      int ka = k + half * 2;
      v2f a, bb;
      a.x = qS[ln * 84 + ka];
      a.y = qS[ln * 84 + ka + 1];
      bb.x = kT[(long)ka * JPAD + j0 + ln];
      bb.y = kT[(long)(ka + 1) * JPAD + j0 + ln];
      acc = wmma4(a, bb, acc);
    }
    int j = j0 + ln;
    int ok = (j < JJ) && (ct[b * JJ + j] >= 0);
#pragma unroll
    for (int r = 0; r < 8; ++r) {
      int m = r + half * 8;
      float val = -3.402823466e38f;
      if (ok) val = acc[r] + cam[((long)b * NN + i0 + m) * JJ + j] * sigma;
      S[m * 324 + j0 + ln] = val;
    }
  }
  __syncthreads();
  {
    int m = tid & 15, seg = tid >> 4, c0 = seg * 20;
    float mx = -3.402823466e38f;
    for (int c = c0; c < c0 + 20; ++c) mx = fmaxf(mx, S[m * 324 + c]);
    red[m * 17 + seg] = mx;
    __syncthreads();
    if (seg == 0) {
      float v = red[m * 17];
      for (int s = 1; s < 16; ++s) v = fmaxf(v, red[m * 17 + s]);
      rowmax[m] = v;
    }
    __syncthreads();
    float rm = rowmax[m], sum = 0.f;
    for (int c = c0; c < c0 + 20; ++c) {
      float e = expf((S[m * 324 + c] - rm) * ATT_SCALE);
      S[m * 324 + c] = e;
      sum += e;
    }
    red[m * 17 + seg] = sum;
    __syncthreads();
    if (seg == 0) {
      float v = 0.f;
      for (int s = 0; s < 16; ++s) v += red[m * 17 + s];
      rowsum[m] = v;
    }
    __syncthreads();
    float inv = 1.0f / rowsum[m];
    for (int c = c0; c < c0 + 20; ++c) S[m * 324 + c] *= inv;
  }
  __syncthreads();
  if (wave < 5) {
    int n0 = wave * 16;
    const float* vb = ws + OFF_V + (long)b * JJ * 640 + h * 80 + n0 + ln;
    v8f acc = zero8();
    for (int k = 0; k < JJ; k += 4) {            // JJ % 4 == 0
      int ka = k + half * 2;
      v2f a, bb;
      a.x = S[ln * 324 + ka];
      a.y = S[ln * 324 + ka + 1];
      bb.x = vb[(long)ka * 640];
      bb.y = vb[(long)(ka + 1) * 640];
      acc = wmma4(a, bb, acc);
    }
    float* ob = ws + OFF_Q + ((long)b * NN + i0) * 640 + h * 80 + n0 + ln;
#pragma unroll
    for (int r = 0; r < 8; ++r) ob[(long)(r + half * 8) * 640] = acc[r];
  }
}

// out = attn_out @ Wo + bo
__global__ __launch_bounds__(256) void k_out(const float* __restrict__ Wo,
                                             const float* __restrict__ bo,
                                             float* __restrict__ out, float* ws) {
  __shared__ float a0[16 * 644];
  int b = blockIdx.x >> 8;
  int i0 = (blockIdx.x & 255) * 16;
  int tid = threadIdx.x, wave = tid >> 5, lane = tid & 31, half = lane >> 4, ln = lane & 15;
  for (int idx = tid; idx < 16 * 160; idx += 256) {
    int r = idx / 160, c4 = (idx - r * 160) * 4;
    *(float4*)(a0 + r * 644 + c4) =
        *(const float4*)(ws + OFF_Q + ((long)b * NN + i0 + r) * 640 + c4);
  }
  __syncthreads();
  for (int t = wave; t < 40; t += 8) {
    int n0 = t * 16;
    v8f acc = mm_ldsA(a0, 644, Wo, 640, n0, 640, lane);
#pragma unroll
    for (int r = 0; r < 8; ++r) {
      int m = r + half * 8;
      out[((long)b * NN + i0 + m) * 640 + n0 + ln] = acc[r] + bo[n0 + ln];
    }
  }
}

extern "C" void kernel_launch(void* const* d_in, const int* in_sizes, int n_in,
                              void* d_out, int out_size, void* d_ws, size_t ws_size,
                              hipStream_t stream) {
  (void)in_sizes; (void)n_in; (void)out_size; (void)ws_size;
  const float* x        = (const float*)d_in[0];
  const float* embs     = (const float*)d_in[1];
  const float* prog     = (const float*)d_in[2];
  const float* cam      = (const float*)d_in[3];
  const float* strength = (const float*)d_in[4];
  const float* tt_emb   = (const float*)d_in[5];
  const float* tt_w1    = (const float*)d_in[6];
  const float* tt_b1    = (const float*)d_in[7];
  const float* tt_w2    = (const float*)d_in[8];
  const float* tt_b2    = (const float*)d_in[9];
  const float* tt_gA    = (const float*)d_in[10];
  const float* tt_gB    = (const float*)d_in[11];
  const float* pe_w1    = (const float*)d_in[12];
  const float* pe_b1    = (const float*)d_in[13];
  const float* pe_w2    = (const float*)d_in[14];
  const float* pe_b2    = (const float*)d_in[15];
  const float* pm_w1    = (const float*)d_in[16];
  const float* pm_b1    = (const float*)d_in[17];
  const float* pm_w2    = (const float*)d_in[18];
  const float* pm_b2    = (const float*)d_in[19];
  const float* pg_gA    = (const float*)d_in[20];
  const float* pg_gB    = (const float*)d_in[21];
  const float* Wq       = (const float*)d_in[22];
  const float* Wk       = (const float*)d_in[23];
  const float* Wv       = (const float*)d_in[24];
  const float* Wo       = (const float*)d_in[25];
  const float* bo       = (const float*)d_in[26];
  const int*   ct       = (const int*)d_in[27];
  float* ws  = (float*)d_ws;
  float* out = (float*)d_out;

  k_init<<<1, 32, 0, stream>>>(ws);
  k_pe<<<BB, 320, 0, stream>>>(prog, pe_w1, pe_b1, pe_w2, pe_b2, ws);
  k_ein<<<(BB * JJ * CDIM / 4 + 255) / 256, 256, 0, stream>>>(embs, ct, tt_emb, ws);
  k_g1<<<BB * 20, 256, 0, stream>>>(tt_w1, tt_b1, ws);
  k_g2<<<BB * 20, 256, 0, stream>>>(embs, tt_w2, tt_b2, tt_gA, tt_gB, ws);
  k_g3<<<BB * 20, 256, 0, stream>>>(Wk, ws);
  k_g4<<<BB * 20, 256, 0, stream>>>(embs, Wv, ws);
  k_qp<<<BB * 256, 256, 0, stream>>>(x, pm_w1, pm_b1, pm_w2, pm_b2, pg_gA, pg_gB, Wq, ws);
  k_gram<<<BB * NH, 256, 0, stream>>>(ct, ws);
  k_scal<<<1, 1, 0, stream>>>(strength, ws);
  k_attn<<<BB * NH * 256, 256, 0, stream>>>(cam, ct, ws);
  k_out<<<BB * 256, 256, 0, stream>>>(Wo, bo, out, ws);
}